// MyModel_87522843561222
// MI455X (gfx1250) — compile-verified
//
#include <hip/hip_runtime.h>

// Batched 2x2 complex Hermitian Cholesky, closed form.
// HBM-streaming kernel: 64 B traffic / matrix, AI ~0.3 FLOP/byte -> memory-bound.
// gfx1250 path: b128 NT loads/stores (single-use data, keep 192MB L2 clean),
// raw v_rsq_f32/v_sqrt_f32 transcendentals (operands bounded in [2,3), so
// 1-2 ulp hardware ops replace the ~30-instruction IEEE refinement sequences
// and keep VALU time safely under the ~11.5us HBM floor).

typedef __attribute__((ext_vector_type(4))) float v4f;

__global__ __launch_bounds__(256) void chol2x2_herm_kernel(
    const v4f* __restrict__ re4,   // real_part: 4 floats per matrix
    const v4f* __restrict__ im4,   // imag_part: 4 floats per matrix
    v4f* __restrict__ out4,        // output: 8 floats (4 complex64) per matrix
    int n)                         // number of matrices
{
    int idx = blockIdx.x * blockDim.x + threadIdx.x;
    if (idx >= n) return;

    // Streaming loads: 128-bit, non-temporal (data used exactly once).
    v4f r = __builtin_nontemporal_load(&re4[idx]); // r00 r01 r10 r11
    v4f m = __builtin_nontemporal_load(&im4[idx]); // i00 i01 i10 i11

    // Hermitian PD construction (diagonal imag parts cancel):
    //   a = r00 + 2, d = r11 + 2 (real diagonal, both in [2,3))
    //   A10 = br - i*bi, with br = (r01+r10)/2, bi = (i01-i10)/2
    float a  = r.x + 2.0f;
    float d  = r.w + 2.0f;
    float br = 0.5f * (r.y + r.z);
    float bi = 0.5f * (m.y - m.z);

    // Cholesky (lower), upper triangle zeroed. With rs = 1/sqrt(a):
    //   L00 = a * rs
    //   L10 = (br - i*bi) * rs
    //   L11 = sqrt(d - (br^2 + bi^2) * rs^2)
    float rs   = __builtin_amdgcn_rsqf(a);        // v_rsq_f32, a in [2,3)
    float l00  = a * rs;
    float l10r = br * rs;
    float l10i = -bi * rs;
    float nrm  = __builtin_fmaf(br, br, bi * bi) * (rs * rs); // = |b|^2 / a
    float l11  = __builtin_amdgcn_sqrtf(d - nrm); // v_sqrt_f32, arg >= ~1.37

    // complex64 row-major interleaved:
    //   [re(L00), im(L00), re(L01), im(L01)] , [re(L10), im(L10), re(L11), im(L11)]
    v4f o0 = {l00, 0.0f, 0.0f, 0.0f};
    v4f o1 = {l10r, l10i, l11, 0.0f};

    __builtin_nontemporal_store(o0, &out4[2 * idx + 0]);
    __builtin_nontemporal_store(o1, &out4[2 * idx + 1]);
}

extern "C" void kernel_launch(void* const* d_in, const int* in_sizes, int n_in,
                              void* d_out, int out_size, void* d_ws, size_t ws_size,
                              hipStream_t stream) {
    (void)n_in; (void)out_size; (void)d_ws; (void)ws_size;

    const v4f* re4 = (const v4f*)d_in[0];   // real_part, float32, B*4 elements
    const v4f* im4 = (const v4f*)d_in[1];   // imag_part, float32, B*4 elements
    v4f* out4      = (v4f*)d_out;           // complex64 out -> 2 float4 per matrix

    int n = in_sizes[0] / 4;                // number of 2x2 matrices (B)
    if (n <= 0) return;

    const int block = 256;                  // 8 wave32s per workgroup
    const int grid  = (n + block - 1) / block;
    chol2x2_herm_kernel<<<grid, block, 0, stream>>>(re4, im4, out4, n);
}